// diffusion_const_field_14439680049268
// MI455X (gfx1250) — compile-verified
//
#include <hip/hip_runtime.h>

#define NSTEPS 100
#define TILE 1024  // edges per block-tile = 256 threads * 4 edges

// ---- CDNA5 async load-to-LDS availability (gfx1250) ------------------------
#if defined(__HIP_DEVICE_COMPILE__) && \
    __has_builtin(__builtin_amdgcn_global_load_async_to_lds_b128) && \
    __has_builtin(__builtin_amdgcn_s_wait_asynccnt)
#define USE_ASYNC_LDS 1
#else
#define USE_ASYNC_LDS 0
#endif

#if USE_ASYNC_LDS
// Parameter types per hipcc diagnostic: v4i in AS(1) (global) / AS(3) (LDS).
typedef int v4i __attribute__((vector_size(4 * sizeof(int))));
typedef __attribute__((address_space(1))) v4i* gb128_t;  // global src
typedef __attribute__((address_space(3))) v4i* lb128_t;  // LDS dst
#endif

__global__ void fill_ones_kernel(float* __restrict__ x, int n) {
  int i = blockIdx.x * blockDim.x + threadIdx.x;
  if (i < n) x[i] = 1.0f;
}

// Compress int64 indices -> int32, fuse DT scaling + row-0 masking rules.
__global__ void prep_edges_kernel(const long long* __restrict__ adj,
                                  const float* __restrict__ pol, int E,
                                  int* __restrict__ di, int* __restrict__ dj,
                                  float* __restrict__ dv) {
  int e = blockIdx.x * blockDim.x + threadIdx.x;
  if (e >= E) return;
  int a = (int)adj[e];
  int b = (int)adj[(size_t)E + e];
  float v = 0.1f * pol[e];
  if (a == 0) v = (b == 0) ? 1.0f : 0.0f;  // row 0: zeroed, except (0,0)->1
  di[e] = a;
  dj[e] = b;
  dv[e] = v;
}

// nxt = diagonal part of M * cur.  Diagonal is (1-DT)=0.9 except M[0][0]=1.
__global__ void init_next_kernel(const float* __restrict__ cur,
                                 float* __restrict__ nxt, int n) {
  int i = blockIdx.x * blockDim.x + threadIdx.x;
  if (i >= n) return;
  nxt[i] = (i == 0) ? cur[0] : (1.0f - 0.1f) * cur[i];
}

// Scatter the off-diagonal edge contributions with native f32 global atomics.
// Edge stream is double-buffered through LDS via async loads when available.
__global__ __launch_bounds__(256) void scatter_kernel(
    const int* __restrict__ di, const int* __restrict__ dj,
    const float* __restrict__ dv, const float* __restrict__ cur,
    float* __restrict__ nxt, int E, int nTiles) {
  const int t = threadIdx.x;
#if USE_ASYNC_LDS
  __shared__ int   s_di[2][TILE];
  __shared__ int   s_dj[2][TILE];
  __shared__ float s_dv[2][TILE];
  const int slot = 4 * t;
  const int cmax = (E - 4) & ~3;  // highest 16B-aligned in-bounds base
  // Clamp global address so every wave issues exactly 3 async ops per tile
  // (uniform ASYNCcnt accounting); tail lanes re-read from global instead.
  auto issue = [&](int tile, int buf) {
    int cbase = tile * TILE + slot;
    if (cbase > cmax) cbase = cmax;
    __builtin_amdgcn_global_load_async_to_lds_b128(
        (gb128_t)(void*)(di + cbase), (lb128_t)(void*)&s_di[buf][slot], 0, 0);
    __builtin_amdgcn_global_load_async_to_lds_b128(
        (gb128_t)(void*)(dj + cbase), (lb128_t)(void*)&s_dj[buf][slot], 0, 0);
    __builtin_amdgcn_global_load_async_to_lds_b128(
        (gb128_t)(void*)(dv + cbase), (lb128_t)(void*)&s_dv[buf][slot], 0, 0);
  };
  int buf = 0;
  if ((int)blockIdx.x < nTiles) issue(blockIdx.x, 0);
  for (int tile = blockIdx.x; tile < nTiles; tile += gridDim.x) {
    int nextT = tile + (int)gridDim.x;
    if (nextT < nTiles) {
      issue(nextT, buf ^ 1);                  // prefetch tile t+1
      __builtin_amdgcn_s_wait_asynccnt(3);    // wait only for tile t's 3 ops
    } else {
      __builtin_amdgcn_s_wait_asynccnt(0);
    }
    int base = tile * TILE + slot;
    if (base + 3 < E) {
      int4   a = *(const int4*)&s_di[buf][slot];   // ds_load_b128
      int4   b = *(const int4*)&s_dj[buf][slot];
      float4 v = *(const float4*)&s_dv[buf][slot];
      unsafeAtomicAdd(&nxt[a.x], v.x * __ldg(&cur[b.x]));
      unsafeAtomicAdd(&nxt[a.y], v.y * __ldg(&cur[b.y]));
      unsafeAtomicAdd(&nxt[a.z], v.z * __ldg(&cur[b.z]));
      unsafeAtomicAdd(&nxt[a.w], v.w * __ldg(&cur[b.w]));
    } else {
      for (int e = base; e < E; ++e)
        unsafeAtomicAdd(&nxt[di[e]], dv[e] * __ldg(&cur[dj[e]]));
    }
    buf ^= 1;
  }
#else
  for (int tile = blockIdx.x; tile < nTiles; tile += gridDim.x) {
    int base = tile * TILE + 4 * t;
    int pf = base + (int)gridDim.x * TILE;
    if (pf + 3 < E) {                      // global_prefetch_b8 next tile
      __builtin_prefetch(di + pf, 0, 1);
      __builtin_prefetch(dj + pf, 0, 1);
      __builtin_prefetch(dv + pf, 0, 1);
    }
    if (base + 3 < E) {
      int4   a = *(const int4*)(di + base);  // global_load_b128
      int4   b = *(const int4*)(dj + base);
      float4 v = *(const float4*)(dv + base);
      unsafeAtomicAdd(&nxt[a.x], v.x * __ldg(&cur[b.x]));
      unsafeAtomicAdd(&nxt[a.y], v.y * __ldg(&cur[b.y]));
      unsafeAtomicAdd(&nxt[a.z], v.z * __ldg(&cur[b.z]));
      unsafeAtomicAdd(&nxt[a.w], v.w * __ldg(&cur[b.w]));
    } else {
      for (int e = base; e < E; ++e)
        unsafeAtomicAdd(&nxt[di[e]], dv[e] * __ldg(&cur[dj[e]]));
    }
  }
#endif
}

// Fallback if workspace too small for int32-compressed edges: stream int64.
__global__ __launch_bounds__(256) void scatter_raw_kernel(
    const long long* __restrict__ adj, const float* __restrict__ pol,
    const float* __restrict__ cur, float* __restrict__ nxt, int E) {
  int stride = gridDim.x * blockDim.x;
  for (int e = blockIdx.x * blockDim.x + threadIdx.x; e < E; e += stride) {
    int a = (int)adj[e];
    int b = (int)adj[(size_t)E + e];
    float v = 0.1f * pol[e];
    if (a == 0) v = (b == 0) ? 1.0f : 0.0f;
    unsafeAtomicAdd(&nxt[a], v * __ldg(&cur[b]));
  }
}

__global__ __launch_bounds__(256) void absmax_partial_kernel(
    const float* __restrict__ x, int n, float* __restrict__ partial) {
  __shared__ float sm[256];
  float m = 0.0f;
  for (int i = 1 + blockIdx.x * blockDim.x + threadIdx.x; i < n;
       i += gridDim.x * blockDim.x)
    m = fmaxf(m, fabsf(x[i]));  // skips index 0 (reference drops spins[0])
  sm[threadIdx.x] = m;
  __syncthreads();
  for (int s = 128; s > 0; s >>= 1) {
    if ((int)threadIdx.x < s)
      sm[threadIdx.x] = fmaxf(sm[threadIdx.x], sm[threadIdx.x + s]);
    __syncthreads();
  }
  if (threadIdx.x == 0) partial[blockIdx.x] = sm[0];
}

__global__ __launch_bounds__(256) void absmax_final_kernel(
    const float* __restrict__ partial, int nb, float* __restrict__ maxv) {
  __shared__ float sm[256];
  float m = 0.0f;
  for (int i = threadIdx.x; i < nb; i += 256) m = fmaxf(m, partial[i]);
  sm[threadIdx.x] = m;
  __syncthreads();
  for (int s = 128; s > 0; s >>= 1) {
    if ((int)threadIdx.x < s)
      sm[threadIdx.x] = fmaxf(sm[threadIdx.x], sm[threadIdx.x + s]);
    __syncthreads();
  }
  if (threadIdx.x == 0) maxv[0] = sm[0];
}

__global__ void normalize_kernel(const float* __restrict__ cur,
                                 const float* __restrict__ maxv,
                                 float* __restrict__ out, int n) {
  int i = blockIdx.x * blockDim.x + threadIdx.x;
  if (i >= n) return;
  out[i] = (i == 0) ? 1.0f : cur[i] / maxv[0];
}

extern "C" void kernel_launch(void* const* d_in, const int* in_sizes, int n_in,
                              void* d_out, int out_size, void* d_ws,
                              size_t ws_size, hipStream_t stream) {
  (void)n_in;
  const long long* adj = (const long long*)d_in[0];
  const float* pol = (const float*)d_in[1];
  const int E = in_sizes[1];
  const int N = out_size;  // == N_1

  char* ws = (char*)d_ws;
  size_t off = 0;
  auto take = [&](size_t bytes) -> void* {
    void* p = (void*)(ws + off);
    off += (bytes + 255) & ~(size_t)255;
    return p;
  };
  float* sa = (float*)take((size_t)N * 4);
  float* sb = (float*)take((size_t)N * 4);
  float* partial = (float*)take(1024 * 4);
  float* maxv = (float*)take(256);
  int* di = (int*)take((size_t)E * 4);
  int* dj = (int*)take((size_t)E * 4);
  float* dv = (float*)take((size_t)E * 4);
  const bool prepped = (off <= ws_size) && (E >= 4);

  const int TB = 256;
  fill_ones_kernel<<<(N + TB - 1) / TB, TB, 0, stream>>>(sa, N);

  const int nTiles = (E + TILE - 1) / TILE;
  const int sBlocks = nTiles < 3072 ? (nTiles > 0 ? nTiles : 1) : 3072;
  if (prepped)
    prep_edges_kernel<<<(E + TB - 1) / TB, TB, 0, stream>>>(adj, pol, E, di,
                                                            dj, dv);

  float* cur = sa;
  float* nxt = sb;
  for (int s = 0; s < NSTEPS; ++s) {
    init_next_kernel<<<(N + TB - 1) / TB, TB, 0, stream>>>(cur, nxt, N);
    if (prepped)
      scatter_kernel<<<sBlocks, TB, 0, stream>>>(di, dj, dv, cur, nxt, E,
                                                 nTiles);
    else
      scatter_raw_kernel<<<sBlocks, TB, 0, stream>>>(adj, pol, cur, nxt, E);
    float* tmp = cur;
    cur = nxt;
    nxt = tmp;
  }

  const int rBlocks = 512;
  absmax_partial_kernel<<<rBlocks, TB, 0, stream>>>(cur, N, partial);
  absmax_final_kernel<<<1, TB, 0, stream>>>(partial, rBlocks, maxv);
  normalize_kernel<<<(N + TB - 1) / TB, TB, 0, stream>>>(cur, maxv,
                                                         (float*)d_out, N);
}